// SLogOrbitalCofactorAntiequivarianceLayer_14851996910246
// MI455X (gfx1250) — compile-verified
//
#include <hip/hip_runtime.h>
#include <math.h>

#define B_DIM 2048
#define NELEC 32
#define D_DIM 128
#define NION  8
#define NSPIN 2
#define NS    16   // NELEC / NSPIN
#define NM    15   // NS - 1 (cofactor submatrix size)

typedef __attribute__((ext_vector_type(2))) float v2f;
typedef __attribute__((ext_vector_type(4))) float v4f;
typedef __attribute__((ext_vector_type(8))) float v8f;

__device__ __forceinline__ float sgnf(float x) {
    return (x > 0.f) ? 1.f : ((x < 0.f) ? -1.f : 0.f);
}

// One block (256 threads = 8 waves) per (spin, batch) pair.
__global__ __launch_bounds__(256)
void slog_cofactor_stage1(const float* __restrict__ eq,
                          const float* __restrict__ rei,
                          const float* __restrict__ W,
                          const float* __restrict__ bias,
                          const float* __restrict__ env_a,
                          const float* __restrict__ env_ion,
                          float* __restrict__ a_sign,
                          float* __restrict__ a_log)
{
    __shared__ float distSh[NS][NION];
    __shared__ float envSh[NS][NS + 1];
    __shared__ float Msh[NS][NS + 1];
    __shared__ float csignSh[NS];
    __shared__ float clogSh[NS];

    const int sb   = blockIdx.x;
    const int s    = sb / B_DIM;
    const int b    = sb - s * B_DIM;
    const int t    = threadIdx.x;
    const int lane = t & 31;
    const int wave = t >> 5;

    // ---- Phase A: distances dist[n][i] = ||r_ei[b, s*16+n, i, :]|| ----
    if (t < NS * NION) {
        const int n = t >> 3, i = t & 7;
        const float* p = rei + (((size_t)b * NELEC + s * NS + n) * NION + i) * 3;
        const float x = p[0], y = p[1], z = p[2];
        distSh[n][i] = sqrtf(x * x + y * y + z * z);
    }
    __syncthreads();

    // ---- Phase B: env[n][o] = sum_i exp(-dist*|env_a|) * env_ion ----
    {
        const int n = t >> 4, o = t & 15;
        const float* ea = env_a   + (size_t)s * NION * NS + o;
        const float* ei = env_ion + (size_t)s * NION * NS + o;
        float acc = 0.f;
        #pragma unroll
        for (int i = 0; i < NION; ++i)
            acc += expf(-distSh[n][i] * fabsf(ea[i * NS])) * ei[i * NS];
        envSh[n][o] = acc;
    }
    __syncthreads();

    // ---- Phase C: 16x16x128 fp32 GEMM via V_WMMA_F32_16X16X4_F32 (wave 0) ----
    if (wave == 0) {
        const int m  = lane & 15;          // A row / B col / C col
        const int kh = (lane >> 4) << 1;   // K sub-offset: 0 (lanes 0-15) or 2 (lanes 16-31)
        const float* Arow = eq + ((size_t)b * NELEC + s * NS + m) * D_DIM;
        const float* Ws   = W + (size_t)s * D_DIM * NS;
        v8f acc = {0.f, 0.f, 0.f, 0.f, 0.f, 0.f, 0.f, 0.f};
        #pragma unroll
        for (int k0 = 0; k0 < D_DIM; k0 += 4) {
            v2f av = *(const v2f*)(Arow + k0 + kh);       // A[m][k0+kh .. +1]
            v2f bv;
            bv.x = Ws[(k0 + kh)     * NS + m];            // B[k0+kh][m]
            bv.y = Ws[(k0 + kh + 1) * NS + m];            // B[k0+kh+1][m]
            acc = __builtin_amdgcn_wmma_f32_16x16x4_f32(
                /*neg_a=*/false, av, /*neg_b=*/false, bv,
                /*c_mod=*/(short)0, acc, /*reuse_a=*/false, /*reuse_b=*/false);
        }
        // C/D layout: VGPR r -> row (r + 8*(lane>=16)), col = lane&15
        const float bb    = bias[(size_t)s * NS + m];
        const int   rbase = (lane >> 4) * 8;
        #pragma unroll
        for (int r = 0; r < 8; ++r)
            Msh[rbase + r][m] = (acc[r] + bb) * envSh[rbase + r][m];
    }
    __syncthreads();

    // ---- Phase D: 16 slogdets of 15x15 cofactor submatrices ----
    // det group k = t>>4 (16 lanes, width-16 shuffles); lane r holds row r.
    const int k = t >> 4;
    const int r = t & 15;
    const int ar = r + (r >= k ? 1 : 0);   // row index into y (skip row k)
    float a[NM];
    #pragma unroll
    for (int c = 0; c < NM; ++c)
        a[c] = (r < NM) ? Msh[ar][c + 1] : 0.f;

    float logacc = 0.f;
    float sgn    = 1.f;
    #pragma unroll
    for (int j = 0; j < NM; ++j) {
        // partial pivot: argmax |a[j]| over rows j..NM-1 (butterfly over 16 lanes)
        float v  = (r >= j && r < NM) ? fabsf(a[j]) : -1.f;
        int   id = r;
        #pragma unroll
        for (int off = 8; off >= 1; off >>= 1) {
            float ov  = __shfl_xor(v, off, 16);
            int   oid = __shfl_xor(id, off, 16);
            if (ov > v || (ov == v && oid < id)) { v = ov; id = oid; }
        }
        const int p = id;
        if (p != j) sgn = -sgn;
        // swap rows j and p
        #pragma unroll
        for (int c = 0; c < NM; ++c) {
            if (c >= j) {
                const float vc = a[c];
                const float vp = __shfl(vc, p, 16);
                const float vj = __shfl(vc, j, 16);
                if (r == j)      a[c] = vp;
                else if (r == p) a[c] = vj;
            }
        }
        const float piv = __shfl(a[j], j, 16);
        logacc += logf(fabsf(piv));
        sgn    *= sgnf(piv);
        const float f = (r > j && r < NM) ? a[j] / piv : 0.f;
        #pragma unroll
        for (int c = j + 1; c < NM; ++c) {
            const float pv = __shfl(a[c], j, 16);
            a[c] -= f * pv;
        }
    }
    if (r == 0) { csignSh[k] = sgn; clogSh[k] = logacc; }
    __syncthreads();

    // ---- Phase E: combine with first column, write per-(s,b,n) scalars ----
    if (t < NS) {
        const float first = Msh[t][0];
        const float osign = sgnf(first);
        const float olog  = logf(fabsf(first));
        const float alt   = (t & 1) ? -1.f : 1.f;
        a_sign[(size_t)sb * NS + t] = osign * csignSh[t] * alt;
        a_log [(size_t)sb * NS + t] = olog + clogSh[t];
    }
}

// Streaming broadcast: out[s][t][b][n][d]; v4f per thread, NT stores.
__global__ __launch_bounds__(256)
void slog_cofactor_stage2(const float* __restrict__ eq,
                          const float* __restrict__ a_sign,
                          const float* __restrict__ a_log,
                          float* __restrict__ out)
{
    const size_t idx = (size_t)blockIdx.x * blockDim.x + threadIdx.x;
    const size_t total = (size_t)NSPIN * B_DIM * NS * (D_DIM / 4);
    if (idx >= total) return;
    const int d4 = (int)(idx & 31);            // 32 float4-groups per row of 128
    const int n  = (int)((idx >> 5) & 15);
    const int b  = (int)((idx >> 9) & 2047);
    const int s  = (int)(idx >> 20);

    const v4f x = *(const v4f*)(eq + (((size_t)b * NELEC + s * NS + n) * D_DIM) + (size_t)d4 * 4);
    const float as = a_sign[((size_t)s * B_DIM + b) * NS + n];
    const float al = a_log [((size_t)s * B_DIM + b) * NS + n];

    v4f osg, olg;
    osg.x = sgnf(x.x) * as;  olg.x = logf(fabsf(x.x)) + al;
    osg.y = sgnf(x.y) * as;  olg.y = logf(fabsf(x.y)) + al;
    osg.z = sgnf(x.z) * as;  olg.z = logf(fabsf(x.z)) + al;
    osg.w = sgnf(x.w) * as;  olg.w = logf(fabsf(x.w)) + al;

    const size_t base0 = ((((size_t)s * 2 + 0) * B_DIM + b) * NS + n) * D_DIM + (size_t)d4 * 4;
    const size_t base1 = base0 + (size_t)B_DIM * NS * D_DIM;
    __builtin_nontemporal_store(osg, (v4f*)(out + base0));
    __builtin_nontemporal_store(olg, (v4f*)(out + base1));
}

extern "C" void kernel_launch(void* const* d_in, const int* in_sizes, int n_in,
                              void* d_out, int out_size, void* d_ws, size_t ws_size,
                              hipStream_t stream) {
    (void)in_sizes; (void)n_in; (void)out_size; (void)ws_size;
    const float* eq      = (const float*)d_in[0];
    const float* rei     = (const float*)d_in[1];
    const float* W       = (const float*)d_in[2];
    const float* bias    = (const float*)d_in[3];
    const float* env_a   = (const float*)d_in[4];
    const float* env_ion = (const float*)d_in[5];
    float* out    = (float*)d_out;
    float* a_sign = (float*)d_ws;                       // NSPIN*B*NS floats
    float* a_log  = a_sign + (size_t)NSPIN * B_DIM * NS;

    slog_cofactor_stage1<<<NSPIN * B_DIM, 256, 0, stream>>>(
        eq, rei, W, bias, env_a, env_ion, a_sign, a_log);

    const size_t total2 = (size_t)NSPIN * B_DIM * NS * (D_DIM / 4);
    slog_cofactor_stage2<<<(unsigned)((total2 + 255) / 256), 256, 0, stream>>>(
        eq, a_sign, a_log, out);
}